// NGCF_22917945491468
// MI455X (gfx1250) — compile-verified
//
#include <hip/hip_runtime.h>

// Problem constants (from reference setup)
#define N_NODES 150000
#define DIM     64
#define LAYERS  3
#define EDGES   1000000
#define BPAIRS  16384

typedef __attribute__((ext_vector_type(2))) float v2f;
typedef __attribute__((ext_vector_type(8))) float v8f;

// ---------------------------------------------------------------------------
// Utility kernels
// ---------------------------------------------------------------------------
__global__ void zero_f32(float* __restrict__ p, int n) {
  int i = blockIdx.x * blockDim.x + threadIdx.x;
  if (i < n) p[i] = 0.0f;
}

__global__ void degree_kernel(const int* __restrict__ edst, float* __restrict__ deg) {
  int i = blockIdx.x * blockDim.x + threadIdx.x;
  if (i < EDGES) atomicAdd(&deg[edst[i]], 1.0f);
}

// deg -> deg^{-1/2} in place (deg entries are >= 0 integers stored as float)
__global__ void dis_kernel(float* __restrict__ deg) {
  int i = blockIdx.x * blockDim.x + threadIdx.x;
  if (i < N_NODES) {
    float d = deg[i];
    deg[i] = (d > 0.0f) ? rsqrtf(d) : 0.0f;
  }
}

// ---------------------------------------------------------------------------
// Fused message GEMM via V_WMMA_F32_16X16X4_F32:
//   msg[n][j] = sum_k x[n][k]*W1[j][k] + sum_k x[n][k]^2 * W2[j][k]
// Treated as one GEMM with K = 128 over A = [x, x*x], B[k][j] = {W1ᵀ; W2ᵀ}.
// Block = 256 threads = 8 waves; each wave computes a 16(M) x 64(N) strip.
// B is staged in LDS in (k-pair, j) float2 layout so each B fragment is one
// ds_load_b64; A fragments are contiguous float2 global loads per the ISA
// 16x4 fp32 A layout (lanes 0-15: K+0,K+1 ; lanes 16-31: K+2,K+3).
// ---------------------------------------------------------------------------
__global__ void __launch_bounds__(256)
gemm_msg_kernel(const float* __restrict__ x, const float* __restrict__ W1,
                const float* __restrict__ W2, float* __restrict__ msg) {
  __shared__ float Bw[2 * DIM * DIM];  // 8192 floats = 32 KB

  // Stage Bcomb[k][j] (k<64 -> W1[j][k], else W2[j][k-64]) pairwise:
  // float2 at [k/2][j] = { Bcomb[k][j], Bcomb[k+1][j] }
  for (int idx = threadIdx.x; idx < 2 * DIM * DIM; idx += 256) {
    int k = idx >> 6;
    int j = idx & 63;
    float v = (k < DIM) ? W1[j * DIM + k] : W2[j * DIM + (k - DIM)];
    Bw[(k >> 1) * (2 * DIM) + j * 2 + (k & 1)] = v;
  }
  __syncthreads();
  const float2* Bp = (const float2*)Bw;

  const int wave = threadIdx.x >> 5;
  const int lane = threadIdx.x & 31;
  const int half = lane >> 4;   // 0: K+0..1, 1: K+2..3 (A/B fragment halves)
  const int r    = lane & 15;
  const int baseM = blockIdx.x * 128 + wave * 16;
  const int row   = baseM + r;
  const int rowc  = (row < N_NODES) ? row : (N_NODES - 1);  // clamp for loads
  const float* __restrict__ xrow = x + (long)rowc * DIM;

  v8f acc0 = {}, acc1 = {}, acc2 = {}, acc3 = {};

  // First 16 k-steps: plain x part (K = 0..63)
#pragma unroll 4
  for (int kk = 0; kk < 16; ++kk) {
    const int klane = kk * 4 + half * 2;
    float2 xa = *(const float2*)(xrow + klane);
    v2f a; a.x = xa.x; a.y = xa.y;
    const int p = kk * 2 + half;
    float2 bw; v2f b;
    bw = Bp[p * DIM + 0 + r];  b.x = bw.x; b.y = bw.y;
    acc0 = __builtin_amdgcn_wmma_f32_16x16x4_f32(false, a, false, b, (short)0, acc0, false, false);
    bw = Bp[p * DIM + 16 + r]; b.x = bw.x; b.y = bw.y;
    acc1 = __builtin_amdgcn_wmma_f32_16x16x4_f32(false, a, false, b, (short)0, acc1, false, false);
    bw = Bp[p * DIM + 32 + r]; b.x = bw.x; b.y = bw.y;
    acc2 = __builtin_amdgcn_wmma_f32_16x16x4_f32(false, a, false, b, (short)0, acc2, false, false);
    bw = Bp[p * DIM + 48 + r]; b.x = bw.x; b.y = bw.y;
    acc3 = __builtin_amdgcn_wmma_f32_16x16x4_f32(false, a, false, b, (short)0, acc3, false, false);
  }
  // Last 16 k-steps: squared part (K = 64..127 -> x[k-64]^2)
#pragma unroll 4
  for (int kk = 16; kk < 32; ++kk) {
    const int klane = kk * 4 + half * 2 - DIM;
    float2 xa = *(const float2*)(xrow + klane);
    xa.x *= xa.x; xa.y *= xa.y;
    v2f a; a.x = xa.x; a.y = xa.y;
    const int p = kk * 2 + half;
    float2 bw; v2f b;
    bw = Bp[p * DIM + 0 + r];  b.x = bw.x; b.y = bw.y;
    acc0 = __builtin_amdgcn_wmma_f32_16x16x4_f32(false, a, false, b, (short)0, acc0, false, false);
    bw = Bp[p * DIM + 16 + r]; b.x = bw.x; b.y = bw.y;
    acc1 = __builtin_amdgcn_wmma_f32_16x16x4_f32(false, a, false, b, (short)0, acc1, false, false);
    bw = Bp[p * DIM + 32 + r]; b.x = bw.x; b.y = bw.y;
    acc2 = __builtin_amdgcn_wmma_f32_16x16x4_f32(false, a, false, b, (short)0, acc2, false, false);
    bw = Bp[p * DIM + 48 + r]; b.x = bw.x; b.y = bw.y;
    acc3 = __builtin_amdgcn_wmma_f32_16x16x4_f32(false, a, false, b, (short)0, acc3, false, false);
  }

  // D layout: VGPR i -> M = i (lanes 0-15) / i+8 (lanes 16-31), N = lane&15
#pragma unroll
  for (int i = 0; i < 8; ++i) {
    const int ro = baseM + i + half * 8;
    if (ro < N_NODES) {
      float* __restrict__ mrow = msg + (long)ro * DIM + r;
      mrow[0]  = acc0[i];
      mrow[16] = acc1[i];
      mrow[32] = acc2[i];
      mrow[48] = acc3[i];
    }
  }
}

// ---------------------------------------------------------------------------
// Weighted SpMM scatter: agg[dst] += dis[dst]*dis[src] * msg[src]
// 16 threads per edge, each handles 4 contiguous dims (float4 gather, 4
// global_atomic_add_f32). msg (38.4 MB) fits in the 192 MB L2.
// ---------------------------------------------------------------------------
__global__ void __launch_bounds__(256)
scatter_kernel(const float* __restrict__ msg, const int* __restrict__ esrc,
               const int* __restrict__ edst, const float* __restrict__ dis,
               float* __restrict__ agg) {
  long t = (long)blockIdx.x * blockDim.x + threadIdx.x;
  long e = t >> 4;
  int  c = (int)(t & 15) * 4;
  if (e < EDGES) {
    int s = esrc[e];
    int d = edst[e];
    float w = dis[s] * dis[d];
    const float4 m = *(const float4*)(msg + (long)s * DIM + c);
    float* __restrict__ ad = agg + (long)d * DIM + c;
    atomicAdd(ad + 0, w * m.x);
    atomicAdd(ad + 1, w * m.y);
    atomicAdd(ad + 2, w * m.z);
    atomicAdd(ad + 3, w * m.w);
  }
}

__global__ void leaky_kernel(float* __restrict__ p, int n) {
  int i = blockIdx.x * blockDim.x + threadIdx.x;
  if (i < n) {
    float v = p[i];
    p[i] = (v > 0.0f) ? v : 0.2f * v;
  }
}

// ---------------------------------------------------------------------------
// Per-layer score contribution: out[b] (+)= dot(x[src_b], x[dst_b]) over 64
// dims. One wave per pair (2 dims/lane as float2), butterfly shuffle reduce.
// concat-dot == sum of per-layer dots, so no 150 MB concat buffer needed.
// ---------------------------------------------------------------------------
__global__ void __launch_bounds__(256)
score_kernel(const float* __restrict__ emb, const int* __restrict__ eli,
             float* __restrict__ out, int accumulate) {
  int wave = threadIdx.x >> 5;
  int lane = threadIdx.x & 31;
  int pair = blockIdx.x * 8 + wave;   // B = 16384 divisible by 8
  int s = eli[pair];
  int d = eli[BPAIRS + pair];
  const float2 a = *(const float2*)(emb + (long)s * DIM + lane * 2);
  const float2 b = *(const float2*)(emb + (long)d * DIM + lane * 2);
  float sum = a.x * b.x + a.y * b.y;
#pragma unroll
  for (int off = 16; off > 0; off >>= 1)
    sum += __shfl_xor(sum, off, 32);
  if (lane == 0) {
    if (accumulate) out[pair] += sum;
    else            out[pair]  = sum;
  }
}

// ---------------------------------------------------------------------------
// Launcher
// ---------------------------------------------------------------------------
extern "C" void kernel_launch(void* const* d_in, const int* in_sizes, int n_in,
                              void* d_out, int out_size, void* d_ws, size_t ws_size,
                              hipStream_t stream) {
  const float* embedding = (const float*)d_in[0];
  const float* W1        = (const float*)d_in[1];  // [L,64,64]
  const float* W2        = (const float*)d_in[2];  // [L,64,64]
  const int*   esrc      = (const int*)d_in[3];
  const int*   edst      = (const int*)d_in[4];
  const int*   eli       = (const int*)d_in[5];    // [2,B] row-major
  float* out = (float*)d_out;

  // Workspace layout (floats): dis[150016] | bufA[N*64] | bufB[N*64] | msg[N*64]
  const long NPAD = 150016;
  float* dis  = (float*)d_ws;
  float* bufA = dis + NPAD;
  float* bufB = bufA + (long)N_NODES * DIM;
  float* msg  = bufB + (long)N_NODES * DIM;

  const int ND = N_NODES * DIM;  // 9,600,000

  // 1) gcn_norm weights
  zero_f32<<<(N_NODES + 255) / 256, 256, 0, stream>>>(dis, N_NODES);
  degree_kernel<<<(EDGES + 255) / 256, 256, 0, stream>>>(edst, dis);
  dis_kernel<<<(N_NODES + 255) / 256, 256, 0, stream>>>(dis);

  // 2) layer-0 score contribution (writes out, defeating workspace poison)
  score_kernel<<<BPAIRS / 8, 256, 0, stream>>>(embedding, eli, out, 0);

  // 3) three propagation layers
  const float* xcur = embedding;
  float* nexts[LAYERS] = {bufA, bufB, bufA};
  for (int l = 0; l < LAYERS; ++l) {
    float* xn = nexts[l];
    gemm_msg_kernel<<<(N_NODES + 127) / 128, 256, 0, stream>>>(
        xcur, W1 + l * DIM * DIM, W2 + l * DIM * DIM, msg);
    zero_f32<<<(ND + 255) / 256, 256, 0, stream>>>(xn, ND);
    scatter_kernel<<<(int)(((long)EDGES * 16 + 255) / 256), 256, 0, stream>>>(
        msg, esrc, edst, dis, xn);
    leaky_kernel<<<(ND + 255) / 256, 256, 0, stream>>>(xn, ND);
    score_kernel<<<BPAIRS / 8, 256, 0, stream>>>(xn, eli, out, 1);
    xcur = xn;
  }
}